// GraphConv_48653389529424
// MI455X (gfx1250) — compile-verified
//
#include <hip/hip_runtime.h>
#include <hip/hip_bf16.h>

#define NN   8192
#define DIN  512
#define DOUT 512

typedef __attribute__((ext_vector_type(2))) float        v2f;
typedef __attribute__((ext_vector_type(8))) float        v8f;
typedef __attribute__((ext_vector_type(4))) unsigned int u32x4;
typedef __attribute__((ext_vector_type(4))) int          i32x4;
typedef __attribute__((ext_vector_type(8))) int          i32x8;

// sched_group_barrier masks
#define SG_WMMA 0x008
#define SG_VMEM_RD 0x020
#define SG_DS_RD 0x100

// D = A(16x4 f32) * B(4x16 f32) + C(16x16 f32), full fp32 precision.
__device__ __forceinline__ v8f wmma_f32(v2f a, v2f b, v8f c) {
  return __builtin_amdgcn_wmma_f32_16x16x4_f32(
      /*neg_a=*/false, a, /*neg_b=*/false, b,
      /*c_mod=*/(short)0, c, /*reuse_a=*/false, /*reuse_b=*/false);
}

// ---------------------------------------------------------------------------
// Tensor Data Mover: DMA `count` contiguous fp32 elements from global memory
// into LDS (count < 65536). D# layout per CDNA5 ISA.
// Issued by one wave; completion via TENSORcnt.
// ---------------------------------------------------------------------------
__device__ __forceinline__ void tdm_load_contig(unsigned lds_off,
                                                const float* gsrc,
                                                unsigned count) {
  unsigned long long ga = (unsigned long long)(const void*)gsrc;
  u32x4 g0;
  g0[0] = 1u;                                   // count=1, user descriptor
  g0[1] = lds_off;                              // lds_addr (bytes)
  g0[2] = (unsigned)(ga & 0xFFFFFFFFu);         // global_addr[31:0]
  g0[3] = (unsigned)((ga >> 32) & 0x1FFFFFFu)   // global_addr[56:32]
        | (2u << 30);                           // type=2 ("image")
  i32x8 g1;
  g1[0] = (int)(2u << 16);                      // data_size=4B; mask/flags=0
  g1[1] = (int)(count << 16);                   // tensor_dim0 = count (lo16)
  g1[2] = (int)((count >> 16) | (1u << 16));    // dim0 hi16; tensor_dim1=1
  g1[3] = (int)(count << 16);                   // tile_dim0 = count
  g1[4] = 1;                                    // tile_dim1 = 1, tile_dim2 = 0
  g1[5] = (int)count;                           // tensor_dim0_stride
  g1[6] = 0;
  g1[7] = 0;
  i32x4 z4 = {0, 0, 0, 0};                      // 2-D tile: groups 2/3 unused
#if defined(__clang_major__) && (__clang_major__ >= 23)
  i32x8 z8 = {0, 0, 0, 0, 0, 0, 0, 0};
  __builtin_amdgcn_tensor_load_to_lds(g0, g1, z4, z4, z8, 0);
#else
  __builtin_amdgcn_tensor_load_to_lds(g0, g1, z4, z4, 0);
#endif
}

// ---------------------------------------------------------------------------
// Kernel 1: h = x @ W (one 16-row stripe per block), inv L2 row norms, and a
// pair-interleaved copy h_pair (per 32-row block: (r,f) -> (r>>1)*1024+f*2+(r&1))
// so that attention's P@h B-fragments become single b64 LDS loads.
// ---------------------------------------------------------------------------
__global__ __launch_bounds__(128) void gemm_norm_kernel(
    const float* __restrict__ x, const float* __restrict__ w,
    float* __restrict__ h, float* __restrict__ h_pair,
    float* __restrict__ inv_norm) {
  __shared__ float xs[16 * DIN];  // 32 KB; x tile, then reused for h tile

  const int tid  = threadIdx.x;
  const int wave = tid >> 5;
  const int lane = tid & 31;
  const int m    = lane & 15;          // row (A) / col (B,C) within tile
  const int kh   = (lane >> 4) * 2;    // K sub-offset for upper lane half
  const int rowbase = blockIdx.x * 16;

  // Stage x rows (contiguous 32 KB) via Tensor Data Mover.
  if (wave == 0) {
    tdm_load_contig((unsigned)(size_t)&xs[0], x + (size_t)rowbase * DIN,
                    16 * DIN);
    __builtin_amdgcn_s_wait_tensorcnt(0);
  }
  __syncthreads();

  // Each wave computes a 16x128 slab = 8 tiles of 16x16, K = 512.
  v8f acc[8] = {};
  for (int k = 0; k < DIN; k += 4) {
    const int kk = k + kh;
    v2f a = *(const v2f*)&xs[m * DIN + kk];  // A[m][kk], A[m][kk+1]
    v2f b[8];
#pragma unroll
    for (int t = 0; t < 8; ++t) {            // batch loads -> one wait
      const int ncol = wave * 128 + t * 16 + m;
      b[t].x = w[(size_t)kk * DOUT + ncol];
      b[t].y = w[(size_t)(kk + 1) * DOUT + ncol];
    }
#pragma unroll
    for (int t = 0; t < 8; ++t) acc[t] = wmma_f32(a, b[t], acc[t]);
    // Pin schedule: A ds-read, 16 weight reads, then 8 back-to-back WMMAs.
    __builtin_amdgcn_sched_group_barrier(SG_DS_RD, 1, 0);
    __builtin_amdgcn_sched_group_barrier(SG_VMEM_RD, 16, 0);
    __builtin_amdgcn_sched_group_barrier(SG_WMMA, 8, 0);
  }
  __syncthreads();

  // Scatter C tiles into LDS as the h tile.
#pragma unroll
  for (int t = 0; t < 8; ++t) {
    const int ncol = wave * 128 + t * 16 + m;
#pragma unroll
    for (int r = 0; r < 8; ++r) {
      const int row = (lane < 16) ? r : (r + 8);
      xs[row * DOUT + ncol] = acc[t][r];
    }
  }
  __syncthreads();

  // Coalesced row-major store of h tile (contiguous 32 KB).
  {
    float4* dst = (float4*)(h + (size_t)rowbase * DOUT);
    const float4* src = (const float4*)xs;
    for (int i = tid; i < 16 * DOUT / 4; i += 128) dst[i] = src[i];
  }

  // Pair-interleaved store: float4 {h[2q][f0], h[2q+1][f0], h[2q][f0+1], h[2q+1][f0+1]}
  {
    const size_t pbase = ((size_t)(rowbase >> 5)) * (32 * DOUT);
    const int qbase = (rowbase & 31) >> 1;     // 0 or 8 within the 32-block
    for (int i = tid; i < 2048; i += 128) {    // 8 q-pairs x 256 f-pairs
      const int ql = i >> 8;                   // local row pair 0..7
      const int f0 = (i & 255) * 2;            // even feature
      v2f e = *(const v2f*)&xs[(2 * ql) * DOUT + f0];      // h[2q][f0..f0+1]
      v2f o = *(const v2f*)&xs[(2 * ql + 1) * DOUT + f0];  // h[2q+1][f0..f0+1]
      float4 v; v.x = e.x; v.y = o.x; v.z = e.y; v.w = o.y;
      *(float4*)&h_pair[pbase + (size_t)(qbase + ql) * 1024 + f0 * 2] = v;
    }
  }

  // Row inverse norms: each wave reduces 4 rows.
  for (int q = 0; q < 4; ++q) {
    const int row = wave * 4 + q;
    float s = 0.f;
    for (int c = lane; c < DOUT; c += 32) {
      const float v = xs[row * DOUT + c];
      s += v * v;
    }
#pragma unroll
    for (int off = 16; off > 0; off >>= 1) s += __shfl_xor(s, off, 32);
    if (lane == 0) inv_norm[rowbase + row] = 1.f / fmaxf(sqrtf(s), 1e-8f);
  }
}

// ---------------------------------------------------------------------------
// Kernel 2: fused  sim -> p = exp(sim-1)*(adj+eps) -> out = (p@h)/rowsum + b
// Br=32 query rows per block, Bc=32 key columns per iteration.
// Fixed softmax shift of 1 is safe: cosine sim <= 1 and log(adj+eps) <= ~0.
// ---------------------------------------------------------------------------
__global__ __launch_bounds__(128) void attn_kernel(
    const float* __restrict__ h, const float* __restrict__ h_pair,
    const float* __restrict__ inv_norm, const float* __restrict__ adj,
    const float* __restrict__ bias, float* __restrict__ out) {
  __shared__ float As[32 * DIN];    // 64 KB: normalized query rows (row-major)
  __shared__ float Hs[32 * DIN];    // 64 KB: key block, row-major (for sim B)
  __shared__ float Hv[32 * DIN];    // 64 KB: key block, pair-interleaved (P@h B)
  __shared__ float Ps[2][16 * 32];  // 4 KB : P tiles (one per row stripe)
  __shared__ float denomS[32];      // per-row softmax denominators

  const int tid  = threadIdx.x;
  const int wave = tid >> 5;
  const int lane = tid & 31;
  const int m    = lane & 15;
  const int kh   = (lane >> 4) * 2;
  const int khh  = lane >> 4;
  const int s    = wave >> 1;   // row stripe (0..1): rows 16*s .. 16*s+15
  const int tc   = wave & 1;    // sim-tile column half / output column half
  const int lrow0 = (lane < 16) ? 0 : 8;
  const int rowbase = blockIdx.x * 32;

  // Stage raw query rows via TDM, then normalize in place.
  if (wave == 0) {
    tdm_load_contig((unsigned)(size_t)&As[0], h + (size_t)rowbase * DIN,
                    32 * DIN);
    __builtin_amdgcn_s_wait_tensorcnt(0);
  }
  if (tid < 32) denomS[tid] = 0.f;
  __syncthreads();
  for (int i = tid; i < 32 * DIN / 4; i += 128) {
    float4 v = ((float4*)As)[i];
    const float inv = inv_norm[rowbase + (i >> 7)];  // i*4/512
    v.x *= inv; v.y *= inv; v.z *= inv; v.w *= inv;
    ((float4*)As)[i] = v;
  }

  v8f  oacc[16] = {};   // 16 rows x 256 cols of out accumulator (fp32)
  float dsum[8] = {};   // per-lane partial denominators

  const int arow_off = (16 * s + m) * DIN;  // A row base in As

  for (int j0 = 0; j0 < NN; j0 += 32) {
    __syncthreads();  // all waves done with previous Hs/Hv/Ps
    if (wave == 0) {  // DMA both layouts of the 64 KB key/value block
      tdm_load_contig((unsigned)(size_t)&Hs[0], h + (size_t)j0 * DIN, 32 * DIN);
      tdm_load_contig((unsigned)(size_t)&Hv[0],
                      h_pair + (size_t)(j0 >> 5) * (32 * DOUT), 32 * DOUT);
      __builtin_amdgcn_s_wait_tensorcnt(0);
    }
    __syncthreads();

    // --- sim tile: rows [rowbase+16s, +16), cols [j0+16tc, +16), K=512 ---
    const int jcol = j0 + tc * 16 + m;      // global key index for this lane
    const float invb = inv_norm[jcol];      // normalize keys on the fly
    const int brow_off = (tc * 16 + m) * DIN;
    v8f c = {};
    for (int k = 0; k < DIN; k += 16) {     // 4 K-steps per trip
      v2f a[4], b[4];
#pragma unroll
      for (int u = 0; u < 4; ++u) {
        const int kk = k + 4 * u + kh;
        a[u] = *(const v2f*)&As[arow_off + kk];
        b[u] = *(const v2f*)&Hs[brow_off + kk];
      }
#pragma unroll
      for (int u = 0; u < 4; ++u) { b[u].x *= invb; b[u].y *= invb; }
#pragma unroll
      for (int u = 0; u < 4; ++u) c = wmma_f32(a[u], b[u], c);
      __builtin_amdgcn_sched_group_barrier(SG_DS_RD, 8, 0);
      __builtin_amdgcn_sched_group_barrier(SG_WMMA, 4, 0);
    }

    // --- p = exp(sim - 1) * (adj + eps); stash P tile in LDS -------------
#pragma unroll
    for (int r = 0; r < 8; ++r) {
      const int lr = lrow0 + r;
      const size_t arow = (size_t)(rowbase + 16 * s + lr);
      const float adjv = adj[arow * NN + jcol];
      __builtin_prefetch(&adj[arow * NN + jcol + 32], 0, 1);  // next j block
      const float p = __expf(c[r] - 1.0f) * (adjv + 1e-8f);
      dsum[r] += p;
      Ps[s][lr * 32 + tc * 16 + m] = p;
    }
    __syncthreads();

    // --- out += P(16x32) @ h_block(32x256-half), K=32 --------------------
    for (int k = 0; k < 32; k += 4) {
      const int kk = k + kh;
      v2f a = *(const v2f*)&Ps[s][m * 32 + kk];
      const int vrow = ((k >> 1) + khh) * 1024;  // pair row (kk>>1) in Hv
      v2f b[16];
#pragma unroll
      for (int t = 0; t < 16; ++t) {  // one b64 per B fragment (interleaved)
        const int col2 = (tc * 256 + t * 16 + m) * 2;
        b[t] = *(const v2f*)&Hv[vrow + col2];  // {H[kk][col], H[kk+1][col]}
      }
#pragma unroll
      for (int t = 0; t < 16; ++t) oacc[t] = wmma_f32(a, b[t], oacc[t]);
      __builtin_amdgcn_sched_group_barrier(SG_DS_RD, 17, 0);
      __builtin_amdgcn_sched_group_barrier(SG_WMMA, 16, 0);
    }
  }

  // Denominators: reduce across the 16 lanes of each half, combine waves.
#pragma unroll
  for (int r = 0; r < 8; ++r) {
    float v = dsum[r];
    v += __shfl_xor(v, 1, 32);
    v += __shfl_xor(v, 2, 32);
    v += __shfl_xor(v, 4, 32);
    v += __shfl_xor(v, 8, 32);
    if (m == 0) atomicAdd(&denomS[16 * s + lrow0 + r], v);
  }
  __syncthreads();

  // Finalize: divide by row denominator, add bias, store fp32.
#pragma unroll
  for (int t = 0; t < 16; ++t) {
    const int col = tc * 256 + t * 16 + m;
    const float bv = bias[col];
#pragma unroll
    for (int r = 0; r < 8; ++r) {
      const int lr = lrow0 + r;
      const float d = denomS[16 * s + lr];
      out[(size_t)(rowbase + 16 * s + lr) * DOUT + col] = oacc[t][r] / d + bv;
    }
  }
}

// ---------------------------------------------------------------------------
extern "C" void kernel_launch(void* const* d_in, const int* in_sizes, int n_in,
                              void* d_out, int out_size, void* d_ws, size_t ws_size,
                              hipStream_t stream) {
  (void)in_sizes; (void)n_in; (void)out_size; (void)ws_size;
  const float* x      = (const float*)d_in[0];   // [8192,512]
  const float* adj    = (const float*)d_in[1];   // [8192,8192]
  const float* weight = (const float*)d_in[2];   // [512,512]
  const float* bias   = (const float*)d_in[3];   // [512]
  float* out = (float*)d_out;                    // [8192,512]

  float* h        = (float*)d_ws;                 // 16 MB row-major
  float* h_pair   = h + (size_t)NN * DOUT;        // 16 MB pair-interleaved
  float* inv_norm = h_pair + (size_t)NN * DOUT;   // 32 KB

  gemm_norm_kernel<<<dim3(NN / 16), dim3(128), 0, stream>>>(
      x, weight, h, h_pair, inv_norm);
  attn_kernel<<<dim3(NN / 32), dim3(128), 0, stream>>>(
      h, h_pair, inv_norm, adj, bias, out);
}